// flowTracker_35046933135755
// MI455X (gfx1250) — compile-verified
//
#include <hip/hip_runtime.h>
#include <float.h>

#define POOL 7
#define KF   98      // 2 channels * 7 * 7
#define KP   100     // padded K (multiple of 4) for WMMA loop
#define FH   2048
#define FW   2048

typedef __attribute__((ext_vector_type(2))) float v2f;
typedef __attribute__((ext_vector_type(8))) float v8f;

// ---------------------------------------------------------------------------
// Kernel 1: per-box adaptive max pool (PyTorch AdaptiveMaxPool2d((7,7)) bins).
// grid.x = N boxes, 128 threads; threads 0..97 each produce one pooled value
// (channel-major flatten c*49 + i*7 + j), threads 98..99 write zero padding.
// ---------------------------------------------------------------------------
__global__ __launch_bounds__(128) void roi_pool_kernel(
    const float* __restrict__ feat,     // [2, 2048, 2048]
    const int*   __restrict__ coords,   // [N, 4] = (y1, x1, y2, x2)
    float*       __restrict__ out)      // [N, KP]
{
  const int box = blockIdx.x;
  const int tid = threadIdx.x;
  if (tid >= KP) return;

  float* orow = out + (size_t)box * KP;
  if (tid >= KF) { orow[tid] = 0.0f; return; }   // zero K padding

  const int c   = tid / 49;
  const int bin = tid % 49;
  const int i   = bin / POOL;
  const int j   = bin % POOL;

  const int y1 = coords[box * 4 + 0];
  const int x1 = coords[box * 4 + 1];
  const int h  = coords[box * 4 + 2] - y1;
  const int w  = coords[box * 4 + 3] - x1;

  // PyTorch adaptive pooling bins: [floor(i*h/P), ceil((i+1)*h/P))
  const int sr = (i * h) / POOL;
  const int er = ((i + 1) * h + POOL - 1) / POOL;
  const int sc = (j * w) / POOL;
  const int ec = ((j + 1) * w + POOL - 1) / POOL;

  const float* base = feat + (size_t)c * FH * FW;
  float m = -FLT_MAX;
  for (int y = y1 + sr; y < y1 + er; ++y) {
    const float* row = base + (size_t)y * FW + x1;
    __builtin_prefetch(row + FW + sc, 0, 0);     // global_prefetch_b8: next row
    for (int x = sc; x < ec; ++x) {
      float v = row[x];
      m = v > m ? v : m;
    }
  }
  orow[tid] = m;
}

// ---------------------------------------------------------------------------
// Kernel 2: FC (feat[N,98] @ fc_w.T[98,4]) + bias + ReLU, then
// out = boxes with columns 2..5 incremented. One wave32 per 16-box tile,
// fp32 WMMA 16x16x4 accumulation over K (24 clean steps + 1 masked tail).
//
// A 16x4 f32 layout: lanes 0-15 -> {K=k, k+1}, lanes 16-31 -> {K=k+2, k+3},
//   M = lane & 15 across lanes.
// B 4x16 f32 layout (mirrored): N = lane & 15 across lanes, same K split.
// C/D 16x16 f32: VGPR v -> M=v (lanes 0-15) / M=v+8 (lanes 16-31), N = lane&15.
//
// All loads are unconditional (clamped-safe addresses); lane predicates are
// applied as value masks (float multiply) so no exec-mask branching appears
// between the loads and the WMMA chain.
// ---------------------------------------------------------------------------
__global__ __launch_bounds__(32) void fc_wmma_kernel(
    const float* __restrict__ featp,    // [N, KP] padded pooled features
    const float* __restrict__ boxes,    // [N, 6]
    const float* __restrict__ fcw,      // [4, 98]
    const float* __restrict__ fcb,      // [4]
    float*       __restrict__ out)      // [N, 6]
{
  const int tile = blockIdx.x;          // 16 boxes per tile
  const int lane = threadIdx.x;         // 0..31 (wave32)
  const int half = lane >> 4;           // 0: K=k,k+1   1: K=k+2,k+3
  const int m    = lane & 15;           // A row (M) / B col (N)

  // A fragment stream: 8-byte aligned (KP and half*2 both even).
  const float* arow = featp + ((size_t)tile * 16 + m) * KP + half * 2;

  // B: row n of fc_w for n<4, else masked to zero. neff keeps address safe.
  const int   neff   = (m < 4) ? m : 0;
  const float bscale = (m < 4) ? 1.0f : 0.0f;
  const float* brow  = fcw + neff * KF + half * 2;   // 8-byte aligned

  v8f acc = {};

  // Main loop: k = 0..92, kk = k + half*2 <= 95 < 98, always in-bounds.
  for (int k = 0; k < 96; k += 4) {
    v2f a = *(const v2f*)(arow + k);
    v2f b = *(const v2f*)(brow + k) * bscale;
    acc = __builtin_amdgcn_wmma_f32_16x16x4_f32(
        false, a, false, b, (short)0, acc, false, false);
  }

  // Tail k = 96: half-0 lanes read kk=96,97 (valid); half-1 lanes would read
  // kk=98,99 -> read a safe dummy address and mask to zero.
  {
    v2f a = *(const v2f*)(arow + 96);                 // featp padded: kk<=99 ok
    const float* btail = fcw + neff * KF + (half ? 0 : 96);
    v2f b = *(const v2f*)(btail) * (half ? 0.0f : bscale);
    acc = __builtin_amdgcn_wmma_f32_16x16x4_f32(
        false, a, false, b, (short)0, acc, false, false);
  }

  const int row0 = tile * 16 + half * 8;
  if (m < 4) {
    // This lane holds output column N=m for rows row0..row0+7 in acc[0..7].
    const float bias = fcb[m];
    for (int v = 0; v < 8; ++v) {
      const int row = row0 + v;
      float d = acc[v] + bias;
      d = d > 0.0f ? d : 0.0f;          // ReLU
      out[row * 6 + 2 + m] = boxes[row * 6 + 2 + m] + d;
    }
  } else if (m < 6) {
    // Lanes 4,5 (and 20,21) copy untouched columns 0,1 for their 8 rows.
    const int col = m - 4;
    for (int v = 0; v < 8; ++v) {
      const int row = row0 + v;
      out[row * 6 + col] = boxes[row * 6 + col];
    }
  }
}

// ---------------------------------------------------------------------------
extern "C" void kernel_launch(void* const* d_in, const int* in_sizes, int n_in,
                              void* d_out, int out_size, void* d_ws, size_t ws_size,
                              hipStream_t stream) {
  const float* feature    = (const float*)d_in[0];  // [1,2,2048,2048]
  const float* boxes      = (const float*)d_in[1];  // [N,6]
  const int*   box_coords = (const int*)  d_in[2];  // [N,4] int32
  const float* fc_w       = (const float*)d_in[3];  // [4,98]
  const float* fc_b       = (const float*)d_in[4];  // [4]
  float*       out        = (float*)d_out;          // [N,6]

  const int N = in_sizes[2] / 4;                    // 1024
  float* featp = (float*)d_ws;                      // [N, KP] scratch

  roi_pool_kernel<<<N, 128, 0, stream>>>(feature, box_coords, featp);
  fc_wmma_kernel<<<N / 16, 32, 0, stream>>>(featp, boxes, fc_w, fc_b, out);
}